// Graph_Attention_Layer_30004641530189
// MI455X (gfx1250) — compile-verified
//
#include <hip/hip_runtime.h>
#include <hip/hip_bf16.h>

typedef __attribute__((ext_vector_type(2))) float v2f;
typedef __attribute__((ext_vector_type(8))) float v8f;

#define D 128  // embedding dim, fixed by the reference

// Order-preserving float <-> u32 encoding so we can use integer atomicMax.
__device__ __forceinline__ unsigned enc_order(float f) {
    unsigned s = __float_as_uint(f);
    return (s & 0x80000000u) ? ~s : (s | 0x80000000u);
}
__device__ __forceinline__ float dec_order(unsigned u) {
    unsigned s = (u & 0x80000000u) ? (u & 0x7fffffffu) : ~u;
    return __uint_as_float(s);
}

// ---------------------------------------------------------------------------
// Pass 1: 16 edges per wave. Gather 16 r0-rows (A, 16x128) and 16 r1-rows
// (B, 128x16 as columns); accumulate 32x V_WMMA_F32_16X16X4_F32; diagonal of
// the 16x16 f32 accumulator = the 16 edge scores. Then atomicMax per node.
//
// A 16x4 f32 layout: lanes 0-15 hold M=0..15 with K={0,1} in the 2 VGPRs,
// lanes 16-31 hold M=0..15 with K={2,3}. B 4x16 mirrors this (rows striped
// across lanes). So for chunk c, lane L fetches 2 consecutive floats at
// row[(L&15)] offset 4c + 2*(L>>4) for BOTH operands: one global_load_b64
// per operand per chunk off a fixed per-lane base pointer.
// ---------------------------------------------------------------------------
__global__ void gat_scores_wmma(const float* __restrict__ embs,
                                const int* __restrict__ ratings,
                                float* __restrict__ scores,
                                unsigned* __restrict__ segmax,
                                int n_edges) {
    const int lane = threadIdx.x & 31;
    const int tile = blockIdx.x * (blockDim.x >> 5) + (threadIdx.x >> 5);
    const int e0 = tile << 4;
    if (e0 >= n_edges) return;            // wave-uniform: EXEC stays all-1s

    int eL = e0 + (lane & 15);
    if (eL >= n_edges) eL = n_edges - 1;  // clamp (keeps EXEC all-1 for WMMA)
    const int r0 = ratings[2 * eL + 0];
    const int r1 = ratings[2 * eL + 1];

    const int sub = (lane >> 4) << 1;     // 0 for lanes 0-15, 2 for 16-31
    const float* pa = embs + (size_t)r0 * D + sub;
    const float* pb = embs + (size_t)r1 * D + sub;

    v8f acc = {};
#pragma unroll
    for (int c = 0; c < D / 4; ++c) {
        v2f a = *(const v2f*)(pa + 4 * c);
        v2f b = *(const v2f*)(pb + 4 * c);
        // 8 args: (neg_a, A, neg_b, B, c_mod, C, reuse_a, reuse_b)
        acc = __builtin_amdgcn_wmma_f32_16x16x4_f32(
            false, a, false, b, (short)0, acc, false, false);
    }

    // Extract diagonal C[i][i].  C layout: VGPR j holds M=j (lanes 0-15,
    // N=lane) and M=j+8 (lanes 16-31, N=lane-16).  So score i lives at
    // (vgpr=i, lane=i) for i<8 and (vgpr=i-8, lane=i+16) for i>=8.
    const int k = lane & 7;
    float v = acc[0];
    v = (k == 1) ? acc[1] : v;
    v = (k == 2) ? acc[2] : v;
    v = (k == 3) ? acc[3] : v;
    v = (k == 4) ? acc[4] : v;
    v = (k == 5) ? acc[5] : v;
    v = (k == 6) ? acc[6] : v;
    v = (k == 7) ? acc[7] : v;
    const int src = (lane < 8) ? lane : lane + 16;
    const float sc = __shfl(v, src, 32);

    if (lane < 16 && (e0 + lane) < n_edges) {
        scores[e0 + lane] = sc;
        atomicMax(&segmax[r0], enc_order(sc));
    }
}

// ---------------------------------------------------------------------------
// Pass 2: ex = exp(score - segmax[r0]); denom[r0] += ex (L2 atomics).
// Overwrites scores[] with ex in place.
// ---------------------------------------------------------------------------
__global__ void gat_softmax_denom(const int* __restrict__ ratings,
                                  const unsigned* __restrict__ segmax,
                                  float* __restrict__ scores,
                                  float* __restrict__ denom,
                                  int n_edges) {
    const int e = blockIdx.x * blockDim.x + threadIdx.x;
    if (e >= n_edges) return;
    const int r0 = ratings[2 * e];
    const float m = dec_order(segmax[r0]);
    const float ex = __expf(scores[e] - m);
    scores[e] = ex;
    atomicAdd(&denom[r0], ex);
}

// ---------------------------------------------------------------------------
// Pass 3: one wave per edge. Coalesced float4 load of embs[r1] (512 B across
// the wave), then contiguous f32 atomic adds into out[r0*128 ..] — each wave
// hits 4 consecutive 128B lines, so the L2 atomic units batch them.
// ---------------------------------------------------------------------------
__global__ void gat_aggregate(const float* __restrict__ embs,
                              const int* __restrict__ ratings,
                              const float* __restrict__ ex,
                              const float* __restrict__ denom,
                              float* __restrict__ out,
                              int n_edges) {
    const int lane = threadIdx.x & 31;
    const int e = blockIdx.x * (blockDim.x >> 5) + (threadIdx.x >> 5);
    if (e >= n_edges) return;
    const int r0 = ratings[2 * e + 0];
    const int r1 = ratings[2 * e + 1];
    const float attn = ex[e] / denom[r0];

    const float4 v = ((const float4*)(embs + (size_t)r1 * D))[lane];
    float* dst = out + (size_t)r0 * D + lane * 4;
    atomicAdd(dst + 0, attn * v.x);
    atomicAdd(dst + 1, attn * v.y);
    atomicAdd(dst + 2, attn * v.z);
    atomicAdd(dst + 3, attn * v.w);
}

// ---------------------------------------------------------------------------
extern "C" void kernel_launch(void* const* d_in, const int* in_sizes, int n_in,
                              void* d_out, int out_size, void* d_ws, size_t ws_size,
                              hipStream_t stream) {
    const float* embs    = (const float*)d_in[0];
    const int*   ratings = (const int*)d_in[1];
    const int node_num = in_sizes[0] / D;   // derive on host (can't read device scalar in capture)
    const int n_edges  = in_sizes[1] / 2;
    float* out = (float*)d_out;

    // Workspace: scores/ex (n_edges f32) | segmax (node_num u32) | denom (node_num f32)
    char* ws = (char*)d_ws;
    float*    scores = (float*)ws;
    unsigned* segmax = (unsigned*)(ws + (size_t)n_edges * 4);
    float*    denom  = (float*)(ws + (size_t)n_edges * 4 + (size_t)node_num * 4);

    // enc_order(x) > 0 for every real score, so 0 is a valid -inf sentinel.
    hipMemsetAsync(d_out, 0, (size_t)out_size * sizeof(float), stream);
    hipMemsetAsync(segmax, 0, (size_t)node_num * 4, stream);
    hipMemsetAsync(denom, 0, (size_t)node_num * 4, stream);

    const int tiles = (n_edges + 15) / 16;            // 16 edges per wave
    dim3 blk(256);                                    // 8 waves per block
    dim3 g1((tiles + 7) / 8);
    gat_scores_wmma<<<g1, blk, 0, stream>>>(embs, ratings, scores, segmax, n_edges);

    dim3 g2((n_edges + 255) / 256);
    gat_softmax_denom<<<g2, blk, 0, stream>>>(ratings, segmax, scores, denom, n_edges);

    dim3 g3((n_edges + 7) / 8);                       // 1 wave per edge
    gat_aggregate<<<g3, blk, 0, stream>>>(embs, ratings, scores, denom, out, n_edges);
}